// GATLSTMCell_2241972929142
// MI455X (gfx1250) — compile-verified
//
#include <hip/hip_runtime.h>
#include <math.h>

// GATv2-LSTM cell for MI455X (gfx1250, wave32).
// N=16000 nodes, E=256000 edges, IN=HID=128, HEADS=2, ED=16, G=4 gates.

#define HIDC   128
#define EDC    16
#define GATES  4

typedef float v2f __attribute__((ext_vector_type(2)));
typedef float v8f __attribute__((ext_vector_type(8)));

__device__ __forceinline__ void fadd_atomic(float* p, float v) {
    unsafeAtomicAdd(p, v);   // native global_atomic_add_f32 (no CAS loop)
}

// ---------------------------------------------------------------------------
// Column sums of edge_attr [E,16] -> easum[16]  (for self-loop fill 'mean')
// ---------------------------------------------------------------------------
__global__ void k_ea_colsum(const float* __restrict__ ea, int E,
                            float* __restrict__ easum) {
    __shared__ float part[256];
    int t = threadIdx.x;
    int k = t & 15;          // column
    int rg = t >> 4;         // row group 0..15
    float s = 0.f;
    for (int e = blockIdx.x * 16 + rg; e < E; e += gridDim.x * 16)
        s += ea[(size_t)e * EDC + k];
    part[t] = s;
    __syncthreads();
    if (t < 16) {
        float tot = 0.f;
        for (int i = t; i < 256; i += 16) tot += part[i];
        fadd_atomic(&easum[t], tot);
    }
}

// ---------------------------------------------------------------------------
// pre[g][n][c] = b_x[g][c] + b_h[g][c]   (both conv biases folded in)
// ---------------------------------------------------------------------------
__global__ void k_init_pre(const float* __restrict__ bx,
                           const float* __restrict__ bh,
                           float* __restrict__ pre, int N) {
    size_t total = (size_t)GATES * N * HIDC;
    size_t stride = (size_t)gridDim.x * blockDim.x;
    for (size_t i = (size_t)blockIdx.x * blockDim.x + threadIdx.x; i < total; i += stride) {
        int c = (int)(i & (HIDC - 1));
        int g = (int)(i / ((size_t)N * HIDC));
        pre[i] = bx[g * HIDC + c] + bh[g * HIDC + c];
    }
}

// ---------------------------------------------------------------------------
// Batched WMMA GEMM: C[b] = A[b] (N x 128) @ W[b] (128 x 128), f32.
// blockIdx.y in [0,16): (mat = y>>2: 0=Wl_x,1=Wr_x,2=Wl_h,3=Wr_h; g = y&3).
// One wave per 16x16 output tile; K-loop of 32 x V_WMMA_F32_16X16X4_F32.
// ---------------------------------------------------------------------------
__global__ void k_gemm(const float* __restrict__ x, const float* __restrict__ h,
                       const float* __restrict__ Wlx, const float* __restrict__ Wrx,
                       const float* __restrict__ Wlh, const float* __restrict__ Wrh,
                       float* __restrict__ XLx, float* __restrict__ XRx,
                       float* __restrict__ XLh, float* __restrict__ XRh, int N) {
    int b   = blockIdx.y;
    int g   = b & 3;
    int mat = b >> 2;
    const float* A = (mat < 2) ? x : h;
    const float* W = (mat == 0 ? Wlx : mat == 1 ? Wrx : mat == 2 ? Wlh : Wrh)
                     + (size_t)g * HIDC * HIDC;
    float* C = (mat == 0 ? XLx : mat == 1 ? XRx : mat == 2 ? XLh : XRh)
               + (size_t)g * N * HIDC;

    int wave = threadIdx.x >> 5;
    int lane = threadIdx.x & 31;
    int Mtiles = N >> 4;
    int tile = blockIdx.x * 4 + wave;        // wave-uniform
    if (tile >= Mtiles * (HIDC / 16)) return; // whole-wave exit; EXEC stays all-1 for WMMA
    int mt = tile >> 3;                      // 8 N-tiles per row of tiles
    int nt = tile & 7;

    int ln = lane & 15;
    int hi = lane >> 4;                      // 0: K=k0..k0+1, 1: K=k0+2..k0+3
    const float* Arow = A + ((size_t)(mt * 16 + ln)) * HIDC;
    const float* Bcol = W + nt * 16 + ln;

    v8f acc = {};
    #pragma unroll 4
    for (int k0 = 0; k0 < HIDC; k0 += 4) {
        int ka = k0 + (hi << 1);
        v2f av, bv;
        av[0] = Arow[ka];
        av[1] = Arow[ka + 1];
        bv[0] = Bcol[(size_t)ka * HIDC];
        bv[1] = Bcol[(size_t)(ka + 1) * HIDC];
        acc = __builtin_amdgcn_wmma_f32_16x16x4_f32(
            /*neg_a=*/false, av, /*neg_b=*/false, bv,
            /*c_mod=*/(short)0, acc, /*reuse_a=*/false, /*reuse_b=*/false);
    }

    // C/D layout: VGPR r -> (M = mt*16 + r + 8*hi, N = nt*16 + ln)
    float* Cc = C + nt * 16 + ln;
    int mb = mt * 16 + hi * 8;
    #pragma unroll
    for (int r = 0; r < 8; ++r)
        Cc[(size_t)(mb + r) * HIDC] = acc[r];
}

// ---------------------------------------------------------------------------
// Pass A (per side): per-edge GATv2 logits -> exp -> segment-sum denominators.
// One wave per edge; 32 lanes cover the 128 channels (4 each, head = lane>>4).
// ee recomputed from edge_attr @ We (LDS-resident weights) to avoid 1.1GB ws.
// exb/den index: e*16 + sideBase + 2*g + head.
// ---------------------------------------------------------------------------
__global__ void k_edge_logits(const int* __restrict__ ei, const float* __restrict__ ea,
                              const float* __restrict__ We, const float* __restrict__ att,
                              const float* __restrict__ XL, const float* __restrict__ XR,
                              const float* __restrict__ easum,
                              float* __restrict__ exb, float* __restrict__ den,
                              int E, int N, int sideBase) {
    __shared__ float sWe[GATES][EDC][HIDC];   // 32 KB
    __shared__ float sAtt[GATES][HIDC];       // 2 KB  ([g][h*64+c])
    for (int i = threadIdx.x; i < GATES * EDC * HIDC; i += blockDim.x)
        ((float*)sWe)[i] = We[i];
    for (int i = threadIdx.x; i < GATES * HIDC; i += blockDim.x)
        ((float*)sAtt)[i] = att[i];
    __syncthreads();

    int lane = threadIdx.x & 31;
    int wave = threadIdx.x >> 5;
    int nw   = blockDim.x >> 5;
    int Etot = E + N;
    float invE = 1.0f / (float)E;
    int head = lane >> 4;
    int c0   = lane << 2;

    for (int e = blockIdx.x * nw + wave; e < Etot; e += gridDim.x * nw) {
        int src, dst;
        float eav;
        if (e < E) {
            src = ei[e];
            dst = ei[E + e];
            eav = ea[(size_t)e * EDC + (lane & 15)];
        } else {
            src = dst = e - E;
            eav = easum[lane & 15] * invE;      // PyG fill_value='mean'
        }
        #pragma unroll
        for (int g = 0; g < GATES; ++g) {
            const float4 xl4 = ((const float4*)(XL + ((size_t)g * N + src) * HIDC))[lane];
            const float4 xr4 = ((const float4*)(XR + ((size_t)g * N + dst) * HIDC))[lane];
            float vv0 = xl4.x + xr4.x, vv1 = xl4.y + xr4.y;
            float vv2 = xl4.z + xr4.z, vv3 = xl4.w + xr4.w;
            #pragma unroll
            for (int k = 0; k < EDC; ++k) {
                float a = __shfl(eav, k, 32);
                const float* w = &sWe[g][k][c0];
                vv0 += a * w[0]; vv1 += a * w[1]; vv2 += a * w[2]; vv3 += a * w[3];
            }
            // leaky_relu(0.2) then dot with att (per head)
            const float* at = &sAtt[g][head * 64 + (c0 & 63)];
            float lg = 0.f;
            float v;
            v = vv0; v = (v > 0.f) ? v : 0.2f * v; lg += v * at[0];
            v = vv1; v = (v > 0.f) ? v : 0.2f * v; lg += v * at[1];
            v = vv2; v = (v > 0.f) ? v : 0.2f * v; lg += v * at[2];
            v = vv3; v = (v > 0.f) ? v : 0.2f * v; lg += v * at[3];
            // reduce across the 16 lanes of this head (xor masks stay in-group)
            lg += __shfl_xor(lg, 1);
            lg += __shfl_xor(lg, 2);
            lg += __shfl_xor(lg, 4);
            lg += __shfl_xor(lg, 8);
            if ((lane & 15) == 0) {
                float ex = expf(lg);        // no max-subtract: identical softmax
                int sub = sideBase + (g << 1) + head;
                exb[((size_t)e << 4) + sub] = ex;
                fadd_atomic(&den[((size_t)dst << 4) + sub], ex);
            }
        }
    }
}

// ---------------------------------------------------------------------------
// Pass B: alpha = ex/den[dst]; pre[g][dst] += alpha * XL_side[g][src].
// Both sides accumulate into the same pre (gx+gh fused).
// ---------------------------------------------------------------------------
__global__ void k_edge_scatter(const int* __restrict__ ei,
                               const float* __restrict__ XLx,
                               const float* __restrict__ XLh,
                               const float* __restrict__ exb,
                               const float* __restrict__ den,
                               float* __restrict__ pre, int E, int N) {
    int lane = threadIdx.x & 31;
    int wave = threadIdx.x >> 5;
    int nw   = blockDim.x >> 5;
    int Etot = E + N;
    int head = lane >> 4;
    int c0   = lane << 2;

    for (int e = blockIdx.x * nw + wave; e < Etot; e += gridDim.x * nw) {
        int src, dst;
        if (e < E) { src = ei[e]; dst = ei[E + e]; }
        else       { src = dst = e - E; }
        #pragma unroll
        for (int combo = 0; combo < 8; ++combo) {      // combo = side*4 + g
            int g = combo & 3;
            const float* XL = (combo < 4) ? XLx : XLh;
            int sub = (combo << 1) + head;             // side*8 + g*2 + head
            float alpha = exb[((size_t)e << 4) + sub] /
                          den[((size_t)dst << 4) + sub];
            const float4 xl4 = ((const float4*)(XL + ((size_t)g * N + src) * HIDC))[lane];
            float* p = pre + ((size_t)g * N + dst) * HIDC + c0;
            fadd_atomic(p + 0, xl4.x * alpha);
            fadd_atomic(p + 1, xl4.y * alpha);
            fadd_atomic(p + 2, xl4.z * alpha);
            fadd_atomic(p + 3, xl4.w * alpha);
        }
    }
}

// ---------------------------------------------------------------------------
// Node update: LSTM gates + LayerNorm. One block (128 threads) per node.
// out = [h_t (N*128) | c_t (N*128)].
// ---------------------------------------------------------------------------
__global__ void k_node(const float* __restrict__ pre, const float* __restrict__ cprev,
                       const float* __restrict__ lnw, const float* __restrict__ lnb,
                       float* __restrict__ out, int N) {
    __shared__ float red[HIDC];
    int n = blockIdx.x;
    int c = threadIdx.x;
    size_t base = (size_t)n * HIDC + c;
    size_t gstr = (size_t)N * HIDC;
    float p0 = pre[base];
    float p1 = pre[gstr + base];
    float p2 = pre[2 * gstr + base];
    float p3 = pre[3 * gstr + base];
    float it = 1.f / (1.f + expf(-p0));
    float ft = 1.f / (1.f + expf(-p1));
    float ot = 1.f / (1.f + expf(-p2));
    float gt = tanhf(p3);
    float ct = ft * cprev[base] + it * gt;
    float ht = ot * tanhf(ct);

    red[c] = ht; __syncthreads();
    for (int s = 64; s > 0; s >>= 1) { if (c < s) red[c] += red[c + s]; __syncthreads(); }
    float mu = red[0] * (1.f / (float)HIDC);
    __syncthreads();
    float d = ht - mu;
    red[c] = d * d; __syncthreads();
    for (int s = 64; s > 0; s >>= 1) { if (c < s) red[c] += red[c + s]; __syncthreads(); }
    float var = red[0] * (1.f / (float)HIDC);

    out[base]        = d * rsqrtf(var + 1e-5f) * lnw[c] + lnb[c];
    out[gstr + base] = ct;
}

// ---------------------------------------------------------------------------
extern "C" void kernel_launch(void* const* d_in, const int* in_sizes, int n_in,
                              void* d_out, int out_size, void* d_ws, size_t ws_size,
                              hipStream_t stream) {
    const float* x     = (const float*)d_in[0];
    const float* hprev = (const float*)d_in[1];
    const float* cprev = (const float*)d_in[2];
    const int*   ei    = (const int*)d_in[3];
    const float* ea    = (const float*)d_in[4];
    const float* Wlx   = (const float*)d_in[5];
    const float* Wrx   = (const float*)d_in[6];
    const float* Wex   = (const float*)d_in[7];
    const float* attx  = (const float*)d_in[8];
    const float* bx    = (const float*)d_in[9];
    const float* Wlh   = (const float*)d_in[10];
    const float* Wrh   = (const float*)d_in[11];
    const float* Weh   = (const float*)d_in[12];
    const float* atth  = (const float*)d_in[13];
    const float* bh    = (const float*)d_in[14];
    const float* lnw   = (const float*)d_in[15];
    const float* lnb   = (const float*)d_in[16];

    int N = in_sizes[0] / HIDC;     // 16000
    int E = in_sizes[3] / 2;        // 256000
    int Etot = E + N;

    // Workspace layout (f32), ~183 MB: all gather targets fit in 192MB L2.
    float* ws  = (float*)d_ws;
    size_t gnh = (size_t)GATES * N * HIDC;
    float* XLx   = ws;
    float* XRx   = XLx + gnh;
    float* XLh   = XRx + gnh;
    float* XRh   = XLh + gnh;
    float* pre   = XRh + gnh;
    float* exb   = pre + gnh;
    float* den   = exb + (size_t)Etot * 16;
    float* easum = den + (size_t)N * 16;

    hipMemsetAsync(den, 0, ((size_t)N * 16 + 16) * sizeof(float), stream);
    k_ea_colsum<<<256, 256, 0, stream>>>(ea, E, easum);
    k_init_pre<<<2048, 256, 0, stream>>>(bx, bh, pre, N);

    int Mtiles = (N + 15) / 16;
    dim3 ggrid((Mtiles * (HIDC / 16) + 3) / 4, 16);
    k_gemm<<<ggrid, 128, 0, stream>>>(x, hprev, Wlx, Wrx, Wlh, Wrh,
                                      XLx, XRx, XLh, XRh, N);

    k_edge_logits<<<1024, 256, 0, stream>>>(ei, ea, Wex, attx, XLx, XRx, easum,
                                            exb, den, E, N, /*sideBase=*/0);
    k_edge_logits<<<1024, 256, 0, stream>>>(ei, ea, Weh, atth, XLh, XRh, easum,
                                            exb, den, E, N, /*sideBase=*/8);
    k_edge_scatter<<<1024, 256, 0, stream>>>(ei, XLx, XLh, exb, den, pre, E, N);
    k_node<<<N, HIDC, 0, stream>>>(pre, cprev, lnw, lnb, (float*)d_out, N);
}